// Net_20358144983204
// MI455X (gfx1250) — compile-verified
//
#include <hip/hip_runtime.h>
#include <math.h>

typedef __attribute__((ext_vector_type(2))) float v2f;
typedef __attribute__((ext_vector_type(8))) float v8f;

#define NEG_SLOPE 0.2f

// ---------- helpers ----------
__device__ __forceinline__ unsigned fkey(float f) {
  unsigned u = __float_as_uint(f);
  return (u & 0x80000000u) ? ~u : (u | 0x80000000u);
}
__device__ __forceinline__ float kdec(unsigned k) {
  return (k & 0x80000000u) ? __uint_as_float(k & 0x7fffffffu) : __uint_as_float(~k);
}
__device__ __forceinline__ float lrelu(float v) { return v > 0.f ? v : NEG_SLOPE * v; }

// ---------- GEMM1: h1[N,64] = x[N,512] @ W1[512,64], WMMA f32 16x16x4 ----------
// block = 256 thr = 8 waves; block tile = 32 rows x 64 cols.
// wave (w&1) -> M subtile, (w>>1) -> N subtile of 16.
__global__ __launch_bounds__(256) void gat_gemm1(const float* __restrict__ x,
                                                 const float* __restrict__ W1,
                                                 float* __restrict__ h1) {
  __shared__ float ldsW[64 * 64];   // one K-chunk of W1, 16KB
  const int tid  = threadIdx.x;
  const int wave = tid >> 5;
  const int lane = tid & 31;
  const int m0 = blockIdx.x * 32 + (wave & 1) * 16;
  const int n0 = (wave >> 1) * 16;
  const int row   = m0 + (lane & 15);
  const int col   = lane & 15;
  const int khalf = (lane >> 4) << 1;   // 0 or 2 (A/B K split across lane halves)

  v8f acc = {};
  for (int kc = 0; kc < 512; kc += 64) {
    // cooperative stage of W1[kc:kc+64, 0:64] (4096 floats) via float4
    const float4* s4 = (const float4*)(W1 + (size_t)kc * 64);
    float4* d4 = (float4*)ldsW;
#pragma unroll
    for (int i = 0; i < 4; ++i) d4[tid + 256 * i] = s4[tid + 256 * i];
    __syncthreads();

#pragma unroll
    for (int k4 = 0; k4 < 64; k4 += 4) {
      v2f a, b;
      const float* ap = x + (size_t)row * 512 + kc + k4 + khalf;
      a.x = ap[0];
      a.y = ap[1];
      const int kb = k4 + khalf;
      b.x = ldsW[kb * 64 + n0 + col];
      b.y = ldsW[(kb + 1) * 64 + n0 + col];
      acc = __builtin_amdgcn_wmma_f32_16x16x4_f32(false, a, false, b,
                                                  (short)0, acc, false, false);
    }
    __syncthreads();
  }
  // C/D layout: VGPR i -> M = i + 8*(lane>=16), N = lane&15
  const int mb = (lane >> 4) << 3;
#pragma unroll
  for (int i = 0; i < 8; ++i)
    h1[(size_t)(m0 + mb + i) * 64 + n0 + col] = acc[i];
}

// ---------- GEMM2: h2[N,16] = act[N,64] @ W2[64,16], one wave / 16 rows ----------
__global__ __launch_bounds__(256) void gat_gemm2(const float* __restrict__ act,
                                                 const float* __restrict__ W2,
                                                 float* __restrict__ h2, int ntiles) {
  const int gwave = (int)((blockIdx.x * (unsigned)blockDim.x + threadIdx.x) >> 5);
  const int lane  = threadIdx.x & 31;
  if (gwave >= ntiles) return;   // wave-uniform guard (EXEC stays all-1s)
  const int m0 = gwave * 16;
  const int row   = m0 + (lane & 15);
  const int col   = lane & 15;
  const int khalf = (lane >> 4) << 1;

  v8f acc = {};
#pragma unroll
  for (int k4 = 0; k4 < 64; k4 += 4) {
    v2f a, b;
    const float* ap = act + (size_t)row * 64 + k4 + khalf;
    a.x = ap[0];
    a.y = ap[1];
    const int kb = k4 + khalf;
    b.x = W2[kb * 16 + col];
    b.y = W2[(kb + 1) * 16 + col];
    acc = __builtin_amdgcn_wmma_f32_16x16x4_f32(false, a, false, b,
                                                (short)0, acc, false, false);
  }
  const int mb = (lane >> 4) << 3;
#pragma unroll
  for (int i = 0; i < 8; ++i)
    h2[(size_t)(m0 + mb + i) * 16 + col] = acc[i];
}

// ---------- per-(node,head) attention logits, layer 1 ----------
__global__ void gat_escore1(const float* __restrict__ h1,
                            const float* __restrict__ asrc,
                            const float* __restrict__ adst,
                            float* __restrict__ es, float* __restrict__ ed, int N) {
  int gid = blockIdx.x * blockDim.x + threadIdx.x;
  if (gid >= N * 8) return;
  int n = gid >> 3, h = gid & 7;
  const float* hp = h1 + (size_t)n * 64 + h * 8;
  float s = 0.f, d = 0.f;
#pragma unroll
  for (int c = 0; c < 8; ++c) {
    s += hp[c] * asrc[h * 8 + c];
    d += hp[c] * adst[h * 8 + c];
  }
  es[gid] = s;
  ed[gid] = d;
}

// ---------- edge passes, layer 1 (8 heads) ----------
__global__ void gat_edge1_max(const int* __restrict__ ei, int E, int ET,
                              const float* __restrict__ es, const float* __restrict__ ed,
                              unsigned* __restrict__ mkey) {
  int gid = blockIdx.x * blockDim.x + threadIdx.x;
  if (gid >= ET * 8) return;
  int e = gid >> 3, h = gid & 7;
  int s = (e < E) ? ei[e] : (e - E);
  int d = (e < E) ? ei[E + e] : (e - E);
  float v = lrelu(es[s * 8 + h] + ed[d * 8 + h]);
  atomicMax(&mkey[d * 8 + h], fkey(v));
}

__global__ void gat_edge1_sum(const int* __restrict__ ei, int E, int ET,
                              const float* __restrict__ es, const float* __restrict__ ed,
                              const unsigned* __restrict__ mkey, float* __restrict__ ssum) {
  int gid = blockIdx.x * blockDim.x + threadIdx.x;
  if (gid >= ET * 8) return;
  int e = gid >> 3, h = gid & 7;
  int s = (e < E) ? ei[e] : (e - E);
  int d = (e < E) ? ei[E + e] : (e - E);
  float v = lrelu(es[s * 8 + h] + ed[d * 8 + h]);
  atomicAdd(&ssum[d * 8 + h], expf(v - kdec(mkey[d * 8 + h])));
}

// one wave per edge, 2 channels / lane (64 channels)
__global__ void gat_edge1_agg(const int* __restrict__ ei, int E, int ET,
                              const float* __restrict__ es, const float* __restrict__ ed,
                              const unsigned* __restrict__ mkey, const float* __restrict__ ssum,
                              const float* __restrict__ h1, float* __restrict__ out) {
  int gid = blockIdx.x * blockDim.x + threadIdx.x;
  int e = gid >> 5;
  if (e >= ET) return;
  int lane = threadIdx.x & 31;
  int s = (e < E) ? ei[e] : (e - E);
  int d = (e < E) ? ei[E + e] : (e - E);
  int ch0 = lane, ch1 = lane + 32;
  int h0 = ch0 >> 3, h1i = ch1 >> 3;
  float v0 = lrelu(es[s * 8 + h0] + ed[d * 8 + h0]);
  float v1 = lrelu(es[s * 8 + h1i] + ed[d * 8 + h1i]);
  float a0 = expf(v0 - kdec(mkey[d * 8 + h0])) / (ssum[d * 8 + h0] + 1e-16f);
  float a1 = expf(v1 - kdec(mkey[d * 8 + h1i])) / (ssum[d * 8 + h1i] + 1e-16f);
  atomicAdd(&out[(size_t)d * 64 + ch0], a0 * h1[(size_t)s * 64 + ch0]);
  atomicAdd(&out[(size_t)d * 64 + ch1], a1 * h1[(size_t)s * 64 + ch1]);
}

// ---------- out1 + b1, then ELU (in place) ----------
__global__ void gat_bias_elu(float* __restrict__ o1, const float* __restrict__ b1, int n64) {
  int gid = blockIdx.x * blockDim.x + threadIdx.x;
  if (gid >= n64) return;
  float v = o1[gid] + b1[gid & 63];
  o1[gid] = v > 0.f ? v : expm1f(v);
}

// ---------- per-node logits, layer 2 (H=1, C=16) ----------
__global__ void gat_escore2(const float* __restrict__ h2,
                            const float* __restrict__ asrc,
                            const float* __restrict__ adst,
                            float* __restrict__ es, float* __restrict__ ed, int N) {
  int n = blockIdx.x * blockDim.x + threadIdx.x;
  if (n >= N) return;
  const float* hp = h2 + (size_t)n * 16;
  float s = 0.f, d = 0.f;
#pragma unroll
  for (int c = 0; c < 16; ++c) {
    s += hp[c] * asrc[c];
    d += hp[c] * adst[c];
  }
  es[n] = s;
  ed[n] = d;
}

__global__ void gat_edge2_max(const int* __restrict__ ei, int E, int ET,
                              const float* __restrict__ es, const float* __restrict__ ed,
                              unsigned* __restrict__ mkey) {
  int e = blockIdx.x * blockDim.x + threadIdx.x;
  if (e >= ET) return;
  int s = (e < E) ? ei[e] : (e - E);
  int d = (e < E) ? ei[E + e] : (e - E);
  atomicMax(&mkey[d], fkey(lrelu(es[s] + ed[d])));
}

__global__ void gat_edge2_sum(const int* __restrict__ ei, int E, int ET,
                              const float* __restrict__ es, const float* __restrict__ ed,
                              const unsigned* __restrict__ mkey, float* __restrict__ ssum) {
  int e = blockIdx.x * blockDim.x + threadIdx.x;
  if (e >= ET) return;
  int s = (e < E) ? ei[e] : (e - E);
  int d = (e < E) ? ei[E + e] : (e - E);
  float v = lrelu(es[s] + ed[d]);
  atomicAdd(&ssum[d], expf(v - kdec(mkey[d])));
}

// 16 threads per edge (one channel each)
__global__ void gat_edge2_agg(const int* __restrict__ ei, int E, int ET,
                              const float* __restrict__ es, const float* __restrict__ ed,
                              const unsigned* __restrict__ mkey, const float* __restrict__ ssum,
                              const float* __restrict__ h2, float* __restrict__ out) {
  int gid = blockIdx.x * blockDim.x + threadIdx.x;
  int e = gid >> 4;
  if (e >= ET) return;
  int c = gid & 15;
  int s = (e < E) ? ei[e] : (e - E);
  int d = (e < E) ? ei[E + e] : (e - E);
  float v = lrelu(es[s] + ed[d]);
  float a = expf(v - kdec(mkey[d])) / (ssum[d] + 1e-16f);
  atomicAdd(&out[(size_t)d * 16 + c], a * h2[(size_t)s * 16 + c]);
}

// ---------- + b2 then log_softmax over 16 classes ----------
__global__ void gat_final(const float* __restrict__ o2, const float* __restrict__ b2,
                          float* __restrict__ out, int N) {
  int n = blockIdx.x * blockDim.x + threadIdx.x;
  if (n >= N) return;
  float v[16];
  float m = -INFINITY;
#pragma unroll
  for (int c = 0; c < 16; ++c) {
    v[c] = o2[(size_t)n * 16 + c] + b2[c];
    m = fmaxf(m, v[c]);
  }
  float sum = 0.f;
#pragma unroll
  for (int c = 0; c < 16; ++c) sum += expf(v[c] - m);
  float l = logf(sum);
#pragma unroll
  for (int c = 0; c < 16; ++c) out[(size_t)n * 16 + c] = v[c] - m - l;
}

extern "C" void kernel_launch(void* const* d_in, const int* in_sizes, int n_in,
                              void* d_out, int out_size, void* d_ws, size_t ws_size,
                              hipStream_t stream) {
  const float* x     = (const float*)d_in[0];
  const int*   eidx  = (const int*)d_in[1];
  const float* W1    = (const float*)d_in[2];
  const float* asrc1 = (const float*)d_in[3];
  const float* adst1 = (const float*)d_in[4];
  const float* b1    = (const float*)d_in[5];
  const float* W2    = (const float*)d_in[6];
  const float* asrc2 = (const float*)d_in[7];
  const float* adst2 = (const float*)d_in[8];
  const float* b2    = (const float*)d_in[9];

  const int N  = in_sizes[0] / 512;   // 100000
  const int E  = in_sizes[1] / 2;     // 3200000
  const int ET = E + N;               // self-loops appended

  // workspace layout (floats)
  float*    h1  = (float*)d_ws;                      // N*64
  float*    es1 = h1 + (size_t)N * 64;               // N*8
  float*    ed1 = es1 + (size_t)N * 8;               // N*8
  unsigned* m1k = (unsigned*)(ed1 + (size_t)N * 8);  // N*8
  float*    s1  = (float*)(m1k + (size_t)N * 8);     // N*8
  float*    o1  = s1 + (size_t)N * 8;                // N*64 (agg -> act, in place)
  float*    h2  = o1 + (size_t)N * 64;               // N*16
  float*    es2 = h2 + (size_t)N * 16;               // N
  float*    ed2 = es2 + N;                           // N
  unsigned* m2k = (unsigned*)(ed2 + N);              // N
  float*    s2  = (float*)(m2k + N);                 // N
  float*    o2  = s2 + N;                            // N*16

  // deterministic re-init of all accumulators (graph-capture safe)
  hipMemsetAsync(m1k, 0, (size_t)N * 8 * 4, stream);
  hipMemsetAsync(s1,  0, (size_t)N * 8 * 4, stream);
  hipMemsetAsync(o1,  0, (size_t)N * 64 * 4, stream);
  hipMemsetAsync(m2k, 0, (size_t)N * 4, stream);
  hipMemsetAsync(s2,  0, (size_t)N * 4, stream);
  hipMemsetAsync(o2,  0, (size_t)N * 16 * 4, stream);

  // ---- layer 1 ----
  gat_gemm1<<<N / 32, 256, 0, stream>>>(x, W1, h1);
  gat_escore1<<<(N * 8 + 255) / 256, 256, 0, stream>>>(h1, asrc1, adst1, es1, ed1, N);
  gat_edge1_max<<<(ET * 8 + 255) / 256, 256, 0, stream>>>(eidx, E, ET, es1, ed1, m1k);
  gat_edge1_sum<<<(ET * 8 + 255) / 256, 256, 0, stream>>>(eidx, E, ET, es1, ed1, m1k, s1);
  gat_edge1_agg<<<(ET + 7) / 8, 256, 0, stream>>>(eidx, E, ET, es1, ed1, m1k, s1, h1, o1);
  gat_bias_elu<<<(N * 64 + 255) / 256, 256, 0, stream>>>(o1, b1, N * 64);

  // ---- layer 2 ----
  const int ntiles = N / 16;
  gat_gemm2<<<(ntiles + 7) / 8, 256, 0, stream>>>(o1, W2, h2, ntiles);
  gat_escore2<<<(N + 255) / 256, 256, 0, stream>>>(h2, asrc2, adst2, es2, ed2, N);
  gat_edge2_max<<<(ET + 255) / 256, 256, 0, stream>>>(eidx, E, ET, es2, ed2, m2k);
  gat_edge2_sum<<<(ET + 255) / 256, 256, 0, stream>>>(eidx, E, ET, es2, ed2, m2k, s2);
  gat_edge2_agg<<<(ET * 16 + 255) / 256, 256, 0, stream>>>(eidx, E, ET, es2, ed2, m2k, s2, h2, o2);
  gat_final<<<(N + 255) / 256, 256, 0, stream>>>(o2, b2, (float*)d_out, N);
}